// MatrixProductState_28192165330990
// MI455X (gfx1250) — compile-verified
//
#include <hip/hip_runtime.h>

// MPS center-orthogonalization sweep, reduced to:
//   G_0 = A0^H A0 (A0 = t_first reshaped 2x512)
//   for j = 0..21:  G <- sum_c T_c^H G T_c          (complex f32 WMMA GEMMs)
//   G = V D V^H (Hermitian Jacobi), sigma = sqrt(D) sorted descending
//   out = diag(sigma) V^H t_last
// f32 WMMA (v_wmma_f32_16x16x4_f32) is required: the 22-step Gram chain
// squares condition numbers; 16-bit matrix paths would destroy small-sigma
// rows. Data (92 MB) is L2-resident on MI455X (192 MB), so the GEMM chain is
// compute/latency bound -> stage panels into LDS with the async global->LDS
// engine (ASYNCcnt), double-buffer, and fold the spin sum into K (K=1024) so
// each step is exactly two GEMM launches.

typedef __attribute__((ext_vector_type(2))) float v2f;
typedef __attribute__((ext_vector_type(4))) float v4f;
typedef __attribute__((ext_vector_type(8))) float v8f;

#define LDM 512
#define PLANE (512 * 512)

// LDS panel geometry: K sliced in KB=16; strides padded (multiple of 4 floats
// to keep 16B-aligned b128 stores) for conflict-free ds_read fragment access.
#define KB 16
#define A_STRIDE 20                            // 16 + pad
#define B_STRIDE 72                            // 64 + pad
#define A_PLANE (64 * A_STRIDE)                // 1280 floats
#define B_PLANE (KB * B_STRIDE)                // 1152 floats
#define BUF_FLOATS (2 * A_PLANE + 2 * B_PLANE) // 4864 floats / buffer

static __device__ __forceinline__ v8f wmma4f32(v2f a, v2f b, v8f c) {
  // V_WMMA_F32_16X16X4_F32: D = A(16x4) * B(4x16) + C(16x16)
  return __builtin_amdgcn_wmma_f32_16x16x4_f32(false, a, false, b, (short)0, c,
                                               false, false);
}

#if __has_builtin(__builtin_amdgcn_global_load_async_to_lds_b128)
#define ASYNC_LDS 1
// Param pointee type (from probe diagnostic): int __attribute__((vector_size(16))).
typedef int v4i_vs __attribute__((vector_size(16)));
typedef __attribute__((address_space(1))) v4i_vs* as1_v4i;
typedef __attribute__((address_space(3))) v4i_vs* as3_v4i;
static __device__ __forceinline__ void copy_b128(const float* g, float* l) {
  // copies 16 bytes per lane from the global address g to the LDS address l, tracked by ASYNCcnt.
  __builtin_amdgcn_global_load_async_to_lds_b128(
      (as1_v4i)(unsigned long long)g,
      (as3_v4i)(unsigned int)(unsigned long long)l, 0, 0);
}
#else
#define ASYNC_LDS 0
static __device__ __forceinline__ void copy_b128(const float* g, float* l) {
  *(v4f*)l = *(const v4f*)g;  // synchronous fallback
}
#endif

// ---------------------------------------------------------------------------
// G_0[d,e] = sum_q conj(a[q,d]) a[q,e], a = t_first (1,2,512,2)
__global__ __launch_bounds__(256) void k_gram_init(const float* __restrict__ tf,
                                                   float* __restrict__ Gr,
                                                   float* __restrict__ Gi) {
  int idx = blockIdx.x * 256 + threadIdx.x;
  int d = idx >> 9, e = idx & 511;
  float ar0 = tf[d * 2], ai0 = tf[d * 2 + 1];
  float br0 = tf[e * 2], bi0 = tf[e * 2 + 1];
  float ar1 = tf[1024 + d * 2], ai1 = tf[1024 + d * 2 + 1];
  float br1 = tf[1024 + e * 2], bi1 = tf[1024 + e * 2 + 1];
  Gr[idx] = ar0 * br0 + ai0 * bi0 + ar1 * br1 + ai1 * bi1;
  Gi[idx] = ar0 * bi0 - ai0 * br0 + ar1 * bi1 - ai1 * br1;
}

// ---------------------------------------------------------------------------
// Deinterleave t_mid[j] (512,2,512,2):
//   T  (stage-1 B): planar [Tr0][Ti0][Tr1][Ti1], each 512x512 (ld 512)
//   TT (stage-2 A): conj-transpose source as ONE 512x1024 plane per part:
//                   TTr[d, c*512+b] at TT, TTi at TT+2*PLANE  (ld 1024)
__global__ __launch_bounds__(256) void k_split(const float* __restrict__ tm,
                                               float* __restrict__ T,
                                               float* __restrict__ TT) {
  int idx = blockIdx.x * 256 + threadIdx.x;
  int b = idx >> 9, d = idx & 511;
#pragma unroll
  for (int c = 0; c < 2; ++c) {
    float re = tm[b * 2048 + c * 1024 + d * 2];
    float im = tm[b * 2048 + c * 1024 + d * 2 + 1];
    T[c * 2 * PLANE + b * LDM + d] = re;
    T[c * 2 * PLANE + PLANE + b * LDM + d] = im;
    TT[(long)d * 1024 + c * 512 + b] = re;
    TT[2 * PLANE + (long)d * 1024 + c * 512 + b] = im;
  }
}

// ---------------------------------------------------------------------------
// Cooperative panel stage: 64xKB A-panel (2 planes) + KBx64 B-panel (2 planes)
// into LDS buffer. 8 b128 transfers per thread per panel, fully coalesced.
static __device__ __forceinline__ void stage_panel(
    const float* __restrict__ Ar, const float* __restrict__ Ai,
    const float* __restrict__ Br, const float* __restrict__ Bi, int M0, int N0,
    int k0, int lda, int ldb, float* buf, int tid) {
#pragma unroll
  for (int pl = 0; pl < 2; ++pl) {
    const float* gA = pl ? Ai : Ar;
    const float* gB = pl ? Bi : Br;
    float* sA = buf + pl * A_PLANE;
    float* sB = buf + 2 * A_PLANE + pl * B_PLANE;
#pragma unroll
    for (int i = 0; i < 2; ++i) {
      int c = tid + i * 128;
      int arow = c >> 2, akc = (c & 3) * 4;  // 64 rows x 4 chunks of 4 floats
      copy_b128(gA + (long)(M0 + arow) * lda + k0 + akc,
                sA + arow * A_STRIDE + akc);
      int brow = c >> 4, bnc = (c & 15) * 4;  // KB rows x 16 chunks
      copy_b128(gB + (long)(k0 + brow) * ldb + N0 + bnc,
                sB + brow * B_STRIDE + bnc);
    }
  }
}

// ---------------------------------------------------------------------------
// Complex GEMM, C(512x512) = [conj?](A(64-row tiles, ld=lda)) * B(ld=ldb),
// K runtime (512 or 1024), LDS double-buffered async staging.
// 4 waves/WG (2x2), 32x32 complex tile per wave. 4 real f32 WMMA accumulators
// per 16x16 subtile: P=ArBr Q=AiBi R=ArBi S=AiBr; Cr = P -/+ Q, Ci = R +/- S.
__global__ __launch_bounds__(128) void k_cgemm(
    const float* __restrict__ ArB, const float* __restrict__ AiB, long aZ,
    const float* __restrict__ BrB, const float* __restrict__ BiB, long bZ,
    float* __restrict__ CrB, float* __restrict__ CiB, long cZ, int lda,
    int ldb, int K, int conjA) {
  const long z = blockIdx.z;
  const float* __restrict__ Ar = ArB + z * aZ;
  const float* __restrict__ Ai = AiB + z * aZ;
  const float* __restrict__ Br = BrB + z * bZ;
  const float* __restrict__ Bi = BiB + z * bZ;
  float* __restrict__ Cr = CrB + z * cZ;
  float* __restrict__ Ci = CiB + z * cZ;

  __shared__ float smem[2 * BUF_FLOATS];  // 38 KB (of 320 KB/WGP)

  const int tid = threadIdx.x;
  const int lane = tid & 31;
  const int wave = tid >> 5;
  const int M0 = blockIdx.y * 64;
  const int N0 = blockIdx.x * 64;
  const int m0w = (wave >> 1) * 32;
  const int n0w = (wave & 1) * 32;
  const int half = lane >> 4;
  const int r16 = lane & 15;

  v8f P[2][2], Q[2][2], R[2][2], S[2][2];
  v8f zero8 = {0.f, 0.f, 0.f, 0.f, 0.f, 0.f, 0.f, 0.f};
#pragma unroll
  for (int mi = 0; mi < 2; ++mi)
#pragma unroll
    for (int ni = 0; ni < 2; ++ni) {
      P[mi][ni] = zero8;
      Q[mi][ni] = zero8;
      R[mi][ni] = zero8;
      S[mi][ni] = zero8;
    }

  const int npanel = K / KB;
  stage_panel(Ar, Ai, Br, Bi, M0, N0, 0, lda, ldb, smem, tid);

  for (int kp = 0; kp < npanel; ++kp) {
    float* cbuf = smem + (kp & 1) * BUF_FLOATS;
    if (kp + 1 < npanel)
      stage_panel(Ar, Ai, Br, Bi, M0, N0, (kp + 1) * KB, lda, ldb,
                  smem + ((kp + 1) & 1) * BUF_FLOATS, tid);
#if ASYNC_LDS
    if (kp + 1 < npanel)
      asm volatile("s_wait_asynccnt 0x8" ::: "memory");  // drain panel kp only
    else
      asm volatile("s_wait_asynccnt 0x0" ::: "memory");
#endif
    __syncthreads();

    const float* sAr = cbuf;
    const float* sAi = cbuf + A_PLANE;
    const float* sBr = cbuf + 2 * A_PLANE;
    const float* sBi = cbuf + 2 * A_PLANE + B_PLANE;
#pragma unroll
    for (int kk = 0; kk < KB; kk += 4) {
      v2f a_r[2], a_i[2], b_r[2], b_i[2];
#pragma unroll
      for (int mi = 0; mi < 2; ++mi) {
        const int arow = m0w + mi * 16 + r16;
        a_r[mi] = *(const v2f*)(sAr + arow * A_STRIDE + kk + 2 * half);
        a_i[mi] = *(const v2f*)(sAi + arow * A_STRIDE + kk + 2 * half);
      }
#pragma unroll
      for (int ni = 0; ni < 2; ++ni) {
        const int bcol = n0w + ni * 16 + r16;
        const int brow = kk + 2 * half;
        b_r[ni].x = sBr[brow * B_STRIDE + bcol];
        b_r[ni].y = sBr[(brow + 1) * B_STRIDE + bcol];
        b_i[ni].x = sBi[brow * B_STRIDE + bcol];
        b_i[ni].y = sBi[(brow + 1) * B_STRIDE + bcol];
      }
#pragma unroll
      for (int mi = 0; mi < 2; ++mi)
#pragma unroll
        for (int ni = 0; ni < 2; ++ni) {
          P[mi][ni] = wmma4f32(a_r[mi], b_r[ni], P[mi][ni]);
          Q[mi][ni] = wmma4f32(a_i[mi], b_i[ni], Q[mi][ni]);
          R[mi][ni] = wmma4f32(a_r[mi], b_i[ni], R[mi][ni]);
          S[mi][ni] = wmma4f32(a_i[mi], b_r[ni], S[mi][ni]);
        }
    }
    __syncthreads();  // all waves done reading before next overwrite
  }

#pragma unroll
  for (int mi = 0; mi < 2; ++mi)
#pragma unroll
    for (int ni = 0; ni < 2; ++ni) {
      v8f cr, ci;
      if (conjA) {
        cr = P[mi][ni] + Q[mi][ni];
        ci = R[mi][ni] - S[mi][ni];
      } else {
        cr = P[mi][ni] - Q[mi][ni];
        ci = R[mi][ni] + S[mi][ni];
      }
      // C/D layout: VGPR v, lane-half h -> M = v + 8h; lane r16 -> N.
      const int row0 = M0 + m0w + mi * 16 + 8 * half;
      const int col = N0 + n0w + ni * 16 + r16;
#pragma unroll
      for (int v = 0; v < 8; ++v) {
        const long idx = (long)(row0 + v) * LDM + col;
        Cr[idx] = cr[v];
        Ci[idx] = ci[v];
      }
    }
}

// ---------------------------------------------------------------------------
// One-workgroup parallel-order Hermitian Jacobi: G <- J^H G J, V <- V J.
// Round-robin (circle) 1-factorization: 256 disjoint pairs/round, 511 rounds.
__global__ __launch_bounds__(1024) void k_jacobi(float* __restrict__ Gr,
                                                 float* __restrict__ Gi,
                                                 float* __restrict__ Vr,
                                                 float* __restrict__ Vi,
                                                 int sweeps) {
  const int tid = threadIdx.x;
  __shared__ int Pp[256], Pq[256];
  __shared__ float Rc[256], Rs[256], Rfr[256], Rfi[256];

  for (int i = tid; i < PLANE; i += 1024) {
    Vr[i] = 0.f;
    Vi[i] = 0.f;
  }
  __threadfence_block();
  __syncthreads();
  if (tid < 512) Vr[tid * LDM + tid] = 1.0f;
  __threadfence_block();
  __syncthreads();

  for (int sw = 0; sw < sweeps; ++sw) {
    for (int round = 0; round < 511; ++round) {
      if (tid < 256) {
        int sA = tid, sB = 511 - tid;
        int p = (sA == 0) ? 0 : 1 + (sA - 1 + round) % 511;
        int q = 1 + (sB - 1 + round) % 511;
        if (p > q) {
          int t = p;
          p = q;
          q = t;
        }
        Pp[tid] = p;
        Pq[tid] = q;
        float a = Gr[p * LDM + p], d = Gr[q * LDM + q];
        float br = Gr[p * LDM + q], bi = Gi[p * LDM + q];
        float h2 = br * br + bi * bi;
        float c = 1.f, s = 0.f, fr = 1.f, fi = 0.f;
        if (h2 > 1e-36f) {
          float h = sqrtf(h2);
          fr = br / h;
          fi = bi / h;
          float tau = (a - d) / (2.f * h);
          float t = ((tau >= 0.f) ? 1.f : -1.f) /
                    (fabsf(tau) + sqrtf(1.f + tau * tau));
          c = rsqrtf(1.f + t * t);
          s = t * c;
        }
        Rc[tid] = c;
        Rs[tid] = s;
        Rfr[tid] = fr;
        Rfi[tid] = fi;
      }
      __threadfence_block();
      __syncthreads();
      // rows: rp' = c*rp + s*phi*rq ; rq' = -s*rp + c*phi*rq
#pragma unroll 1
      for (int it = 0; it < 128; ++it) {
        int u = tid + it * 1024;
        int pr = u >> 9, k = u & 511;
        int p = Pp[pr], q = Pq[pr];
        float c = Rc[pr], s = Rs[pr], fr = Rfr[pr], fi = Rfi[pr];
        long ip = (long)p * LDM + k, iq = (long)q * LDM + k;
        float xr = Gr[ip], xi = Gi[ip], yr = Gr[iq], yi = Gi[iq];
        float tqr = fr * yr - fi * yi;
        float tqi = fr * yi + fi * yr;
        Gr[ip] = c * xr + s * tqr;
        Gi[ip] = c * xi + s * tqi;
        Gr[iq] = -s * xr + c * tqr;
        Gi[iq] = -s * xi + c * tqi;
      }
      __threadfence_block();
      __syncthreads();
      // cols (phi-bar), and V columns identically
#pragma unroll 1
      for (int it = 0; it < 128; ++it) {
        int u = tid + it * 1024;
        int pr = u >> 9, k = u & 511;
        int p = Pp[pr], q = Pq[pr];
        float c = Rc[pr], s = Rs[pr], fr = Rfr[pr], fi = -Rfi[pr];
        long ip = (long)k * LDM + p, iq = (long)k * LDM + q;
        {
          float xr = Gr[ip], xi = Gi[ip], yr = Gr[iq], yi = Gi[iq];
          float tqr = fr * yr - fi * yi;
          float tqi = fr * yi + fi * yr;
          Gr[ip] = c * xr + s * tqr;
          Gi[ip] = c * xi + s * tqi;
          Gr[iq] = -s * xr + c * tqr;
          Gi[iq] = -s * xi + c * tqi;
        }
        {
          float xr = Vr[ip], xi = Vi[ip], yr = Vr[iq], yi = Vi[iq];
          float tqr = fr * yr - fi * yi;
          float tqi = fr * yi + fi * yr;
          Vr[ip] = c * xr + s * tqr;
          Vi[ip] = c * xi + s * tqi;
          Vr[iq] = -s * xr + c * tqr;
          Vi[iq] = -s * xi + c * tqi;
        }
      }
      __threadfence_block();
      __syncthreads();
    }
  }
}

// ---------------------------------------------------------------------------
// Rank eigenvalues (diag of G) descending -> perm (SVD orders sigma desc).
__global__ __launch_bounds__(512) void k_sort(const float* __restrict__ Gr,
                                              int* __restrict__ perm) {
  __shared__ float lam[512];
  int t = threadIdx.x;
  lam[t] = Gr[t * (LDM + 1)];
  __syncthreads();
  float v = lam[t];
  int rank = 0;
  for (int e = 0; e < 512; ++e) {
    float w = lam[e];
    rank += (w > v) || (w == v && e < t);
  }
  perm[rank] = t;
}

// ---------------------------------------------------------------------------
// out[a,c] = sigma_a * sum_b conj(V[b, perm[a]]) * t_last[b,c]
__global__ __launch_bounds__(512) void k_finalize(
    const float* __restrict__ Vr, const float* __restrict__ Vi,
    const float* __restrict__ Gr, const int* __restrict__ perm,
    const float* __restrict__ tl, float* __restrict__ out) {
  int a = threadIdx.x;
  int da = perm[a];
  float lam = Gr[da * (LDM + 1)];
  float sig = sqrtf(fmaxf(lam, 0.f));
#pragma unroll
  for (int c = 0; c < 2; ++c) {
    float accr = 0.f, acci = 0.f;
    for (int b = 0; b < 512; ++b) {
      float vr = Vr[b * LDM + da], vi = Vi[b * LDM + da];
      float tr = tl[b * 4 + c * 2], ti = tl[b * 4 + c * 2 + 1];
      accr += vr * tr + vi * ti;
      acci += vr * ti - vi * tr;
    }
    out[a * 4 + c * 2] = sig * accr;
    out[a * 4 + c * 2 + 1] = sig * acci;
  }
}

// ---------------------------------------------------------------------------
extern "C" void kernel_launch(void* const* d_in, const int* in_sizes, int n_in,
                              void* d_out, int out_size, void* d_ws,
                              size_t ws_size, hipStream_t stream) {
  (void)in_sizes;
  (void)n_in;
  (void)out_size;
  (void)ws_size;
  const float* t_first = (const float*)d_in[0];
  const float* t_mid = (const float*)d_in[1];
  const float* t_last = (const float*)d_in[2];
  float* out = (float*)d_out;

  const size_t P = PLANE;
  float* ws = (float*)d_ws;
  // ws layout (floats):
  //   G0(2P) G1(2P)
  //   T  (4P): [Tr0][Ti0][Tr1][Ti1]                 (stage-1 B, ld 512)
  //   TT (4P): [TTr 512x1024][TTi 512x1024]          (stage-2 A, ld 1024)
  //   M  (4P): [Mr0][Mr1][Mi0][Mi1]                  (1024x512 stack, ld 512)
  //   V  (2P): [Vr][Vi]   perm(512 ints)
  float* G[2] = {ws, ws + 2 * P};
  float* T = ws + 4 * P;
  float* TT = ws + 8 * P;
  float* M = ws + 12 * P;
  float* Vr = ws + 16 * P;
  float* Vi = ws + 17 * P;
  int* perm = (int*)(ws + 18 * P);

  int cur = 0;
  k_gram_init<<<dim3(1024), dim3(256), 0, stream>>>(t_first, G[cur],
                                                    G[cur] + P);

  for (int j = 0; j < 22; ++j) {
    k_split<<<dim3(1024), dim3(256), 0, stream>>>(t_mid + (size_t)j * 4 * P, T,
                                                  TT);
    // stage1: M_c = G * T_c   (z = c; C planes land as [Mr0][Mr1][Mi0][Mi1])
    k_cgemm<<<dim3(8, 8, 2), dim3(128), 0, stream>>>(
        G[cur], G[cur] + P, 0L, T, T + P, (long)(2 * P), M, M + 2 * P, (long)P,
        512, 512, 512, 0);
    // stage2: G' = conj(TT) * [M0;M1]  -- spin sum folded into K=1024
    int nxt = cur ^ 1;
    k_cgemm<<<dim3(8, 8, 1), dim3(128), 0, stream>>>(
        TT, TT + 2 * P, 0L, M, M + 2 * P, 0L, G[nxt], G[nxt] + P, 0L, 1024,
        512, 1024, 1);
    cur = nxt;
  }

  k_jacobi<<<dim3(1), dim3(1024), 0, stream>>>(G[cur], G[cur] + P, Vr, Vi, 6);
  k_sort<<<dim3(1), dim3(512), 0, stream>>>(G[cur], perm);
  k_finalize<<<dim3(1), dim3(512), 0, stream>>>(Vr, Vi, G[cur], perm, t_last,
                                                out);
}